// ModelNew_17514876633363
// MI455X (gfx1250) — compile-verified
//
#include <hip/hip_runtime.h>

// Mamba-2 SSD chunked scan for gfx1250, FP32 WMMA (V_WMMA_F32_16X16X4_F32).
// Shapes: b=8, s=4096, h=16, p=64, n=16, chunk l=64, chunks c=64.

typedef float v2f __attribute__((ext_vector_type(2)));
typedef float v8f __attribute__((ext_vector_type(8)));

#define WMMA_F32(a, b, c) \
  __builtin_amdgcn_wmma_f32_16x16x4_f32(false, (a), false, (b), (short)0, (c), false, false)

#define NB   8
#define NS   4096
#define NH   16
#define NP   64
#define NN   16
#define CL   64      // chunk length
#define NC   64      // chunks = NS/CL
#define TILES (NB*NC*NH)          // 8192
#define STATE_ELEMS (NN*NP)       // 1024 floats per (b,c,h) state

// ---------------------------------------------------------------------------
// Kernel 1: per-(b,c,h) tile. 4 waves (128 threads). Computes Y_diag and the
// chunk state  S[n][p] = sum_l B[l,n] * exp(cs63-cs[l]) * X[l,p], plus a_total.
// ---------------------------------------------------------------------------
__global__ __launch_bounds__(128) void ssd_chunk_kernel(
    const float* __restrict__ Xg, const float* __restrict__ Ag,
    const float* __restrict__ Bg, const float* __restrict__ Cg,
    float* __restrict__ Yg, float* __restrict__ Sg, float* __restrict__ Atot)
{
  __shared__ __align__(16) float sX[CL * NP];   // X[l][p]
  __shared__ __align__(16) float sB[CL * NN];   // B[l][n], later B*w in place
  __shared__ __align__(16) float sC[CL * NN];   // C[l][n]
  __shared__ __align__(16) float sG[CL * CL];   // masked decay-weighted C*B^T
  __shared__ __align__(16) float sCS[CL];       // inclusive cumsum of A
  __shared__ __align__(16) float sW[CL];        // exp(cs63 - cs[l]) (temp: A)

  const int tid  = threadIdx.x;
  const int lane = tid & 31;
  const int wv   = tid >> 5;          // wave 0..3
  const int bid  = blockIdx.x;        // b*1024 + c*16 + h
  const int h    = bid & 15;
  const int c    = (bid >> 4) & 63;
  const int b    = bid >> 10;

  const long srow  = (long)(b * NS + c * CL);
  const long xbase = (srow * NH + h) * NP;      // stride over s: NH*NP = 1024
  const long bbase = (srow * NH + h) * NN;      // stride over s: NH*NN = 256
  const long abase = srow * NH + h;             // stride over s: NH = 16

  // ---- stage X, B, C into LDS (float4) ----
  for (int i = tid; i < (CL * NP) / 4; i += 128) {       // 1024 float4s
    int l = i >> 4, p4 = (i & 15) << 2;
    *(float4*)&sX[l * NP + p4] = *(const float4*)&Xg[xbase + (long)l * 1024 + p4];
  }
  for (int i = tid; i < (CL * NN) / 4; i += 128) {       // 256 float4s
    int l = i >> 2, n4 = (i & 3) << 2;
    *(float4*)&sB[l * NN + n4] = *(const float4*)&Bg[bbase + (long)l * 256 + n4];
    *(float4*)&sC[l * NN + n4] = *(const float4*)&Cg[bbase + (long)l * 256 + n4];
  }
  if (tid < CL) sW[tid] = Ag[abase + (long)tid * NH];    // raw A into sW (temp)
  __syncthreads();
  if (tid == 0) {                                        // serial inclusive cumsum
    float run = 0.f;
    for (int i = 0; i < CL; ++i) { run += sW[i]; sCS[i] = run; }
  }
  __syncthreads();
  const float a_total = sCS[CL - 1];
  if (tid < CL) sW[tid] = __expf(a_total - sCS[tid]);    // decay_states
  if (tid == 0) Atot[bid] = a_total;
  __syncthreads();

  const int hi   = lane >> 4;      // 0: lanes 0-15, 1: lanes 16-31
  const int lm   = lane & 15;
  const int koff = hi * 2;         // K offset for this half-wave
  const int mt   = wv;             // this wave's M-tile row

  // per-lane row cumsums for this wave's 8 output rows (branchless mask prep)
  float csr[8];
  {
    float4 r0 = *(const float4*)&sCS[mt * 16 + hi * 8];
    float4 r1 = *(const float4*)&sCS[mt * 16 + hi * 8 + 4];
    csr[0] = r0.x; csr[1] = r0.y; csr[2] = r0.z; csr[3] = r0.w;
    csr[4] = r1.x; csr[5] = r1.y; csr[6] = r1.z; csr[7] = r1.w;
  }

  // ---- G = (C * B^T) .* exp(cs[i]-cs[j]) tril mask (branchless) ----
  for (int nt = 0; nt < 4; ++nt) {
    v8f acc = {};
    for (int kk = 0; kk < 4; ++kk) {
      int k0 = kk * 4 + koff;
      v2f af, bf;
      af.x = sC[(mt * 16 + lm) * NN + k0];
      af.y = sC[(mt * 16 + lm) * NN + k0 + 1];
      bf.x = sB[(nt * 16 + lm) * NN + k0];       // B^T[k][j] = B[j][k]
      bf.y = sB[(nt * 16 + lm) * NN + k0 + 1];
      acc = WMMA_F32(af, bf, acc);
    }
    const int   col    = nt * 16 + lm;
    const float cs_col = sCS[col];
    for (int r = 0; r < 8; ++r) {
      int   row = mt * 16 + r + hi * 8;
      float e   = __expf(csr[r] - cs_col);       // unconditional exp
      float g   = (col <= row) ? acc[r] * e : 0.f;  // v_cndmask, no branch
      sG[row * CL + col] = g;
    }
  }
  __syncthreads();

  // in-place: sB[l][n] *= exp(cs63 - cs[l])  (B dead for G now; feeds states)
  for (int i = tid; i < CL * NN; i += 128) sB[i] *= sW[i >> 4];
  __syncthreads();

  // ---- Y_diag = G * X ----
  for (int pt = 0; pt < 4; ++pt) {
    v8f acc = {};
    for (int kk = 0; kk < 16; ++kk) {
      int k0 = kk * 4 + koff;
      v2f af, bf;
      af.x = sG[(mt * 16 + lm) * CL + k0];
      af.y = sG[(mt * 16 + lm) * CL + k0 + 1];
      bf.x = sX[k0 * NP + pt * 16 + lm];
      bf.y = sX[(k0 + 1) * NP + pt * 16 + lm];
      acc = WMMA_F32(af, bf, acc);
    }
    for (int r = 0; r < 8; ++r) {
      int row = mt * 16 + r + hi * 8;
      Yg[xbase + (long)row * 1024 + pt * 16 + lm] = acc[r];
    }
  }

  // ---- chunk state: S[n][p] = sum_l (B*w)[l,n]*X[l,p]; wave w does pt=w ----
  {
    const int pt = wv;
    v8f acc = {};
    for (int kk = 0; kk < 16; ++kk) {
      int k0 = kk * 4 + koff;
      v2f af, bf;
      af.x = sB[k0 * NN + lm];                   // A[m=n][k=l] = (B*w)[l][n]
      af.y = sB[(k0 + 1) * NN + lm];
      bf.x = sX[k0 * NP + pt * 16 + lm];
      bf.y = sX[(k0 + 1) * NP + pt * 16 + lm];
      acc = WMMA_F32(af, bf, acc);
    }
    float* Sb = Sg + (long)bid * STATE_ELEMS;
    for (int r = 0; r < 8; ++r) {
      int nrow = r + hi * 8;
      Sb[nrow * NP + pt * 16 + lm] = acc[r];
    }
  }
}

// ---------------------------------------------------------------------------
// Kernel 2: sequential cross-chunk scan per (b,h). Overwrites the workspace
// in place with the state ENTERING each chunk:  R_{c+1} = exp(a_c)*R_c + S_c.
// ---------------------------------------------------------------------------
__global__ __launch_bounds__(256) void ssd_scan_kernel(
    float* __restrict__ Sg, const float* __restrict__ Atot)
{
  const int bh = blockIdx.x;          // b*16 + h
  const int b  = bh >> 4;
  const int h  = bh & 15;
  const int e  = threadIdx.x * 4;     // this thread's 4 state elements
  float4 carry = make_float4(0.f, 0.f, 0.f, 0.f);
  for (int c = 0; c < NC; ++c) {
    const int  bid  = b * 1024 + c * 16 + h;
    const long base = (long)bid * STATE_ELEMS + e;
    float4 st = *(float4*)&Sg[base];
    *(float4*)&Sg[base] = carry;                 // state entering chunk c
    float d = __expf(Atot[bid]);
    carry.x = d * carry.x + st.x;
    carry.y = d * carry.y + st.y;
    carry.z = d * carry.z + st.z;
    carry.w = d * carry.w + st.w;
  }
}

// ---------------------------------------------------------------------------
// Kernel 3: Y += exp(cs[l]) * (C * S_prev). 4 waves per tile, wave w = M-tile w.
// ---------------------------------------------------------------------------
__global__ __launch_bounds__(128) void ssd_off_kernel(
    const float* __restrict__ Ag, const float* __restrict__ Cg,
    const float* __restrict__ Sg, float* __restrict__ Yg)
{
  __shared__ __align__(16) float sC[CL * NN];
  __shared__ __align__(16) float sS[NN * NP];   // S_prev[n][p]
  __shared__ __align__(16) float sCS[CL];

  const int tid  = threadIdx.x;
  const int lane = tid & 31;
  const int wv   = tid >> 5;
  const int bid  = blockIdx.x;
  const int h    = bid & 15;
  const int c    = (bid >> 4) & 63;
  const int b    = bid >> 10;

  const long srow  = (long)(b * NS + c * CL);
  const long xbase = (srow * NH + h) * NP;
  const long bbase = (srow * NH + h) * NN;
  const long abase = srow * NH + h;

  for (int i = tid; i < (CL * NN) / 4; i += 128) {
    int l = i >> 2, n4 = (i & 3) << 2;
    *(float4*)&sC[l * NN + n4] = *(const float4*)&Cg[bbase + (long)l * 256 + n4];
  }
  for (int i = tid; i < STATE_ELEMS / 4; i += 128) {
    *(float4*)&sS[i * 4] = *(const float4*)&Sg[(long)bid * STATE_ELEMS + i * 4];
  }
  if (tid < CL) sCS[tid] = Ag[abase + (long)tid * NH];
  __syncthreads();
  if (tid == 0) {
    float run = 0.f;
    for (int i = 0; i < CL; ++i) { run += sCS[i]; sCS[i] = run; }
  }
  __syncthreads();

  const int hi   = lane >> 4;
  const int lm   = lane & 15;
  const int koff = hi * 2;
  const int mt   = wv;

  // per-lane exp(cs[row]) for this wave's 8 rows
  float dec[8];
  {
    float4 r0 = *(const float4*)&sCS[mt * 16 + hi * 8];
    float4 r1 = *(const float4*)&sCS[mt * 16 + hi * 8 + 4];
    dec[0] = __expf(r0.x); dec[1] = __expf(r0.y);
    dec[2] = __expf(r0.z); dec[3] = __expf(r0.w);
    dec[4] = __expf(r1.x); dec[5] = __expf(r1.y);
    dec[6] = __expf(r1.z); dec[7] = __expf(r1.w);
  }

  for (int pt = 0; pt < 4; ++pt) {
    v8f acc = {};
    for (int kk = 0; kk < 4; ++kk) {
      int k0 = kk * 4 + koff;
      v2f af, bf;
      af.x = sC[(mt * 16 + lm) * NN + k0];
      af.y = sC[(mt * 16 + lm) * NN + k0 + 1];
      bf.x = sS[k0 * NP + pt * 16 + lm];
      bf.y = sS[(k0 + 1) * NP + pt * 16 + lm];
      acc = WMMA_F32(af, bf, acc);
    }
    for (int r = 0; r < 8; ++r) {
      int row = mt * 16 + r + hi * 8;
      long off = xbase + (long)row * 1024 + pt * 16 + lm;
      Yg[off] = Yg[off] + dec[r] * acc[r];
    }
  }
}

// ---------------------------------------------------------------------------
extern "C" void kernel_launch(void* const* d_in, const int* in_sizes, int n_in,
                              void* d_out, int out_size, void* d_ws, size_t ws_size,
                              hipStream_t stream) {
  const float* X = (const float*)d_in[0];  // [8,4096,16,64]
  const float* A = (const float*)d_in[1];  // [8,4096,16]
  const float* B = (const float*)d_in[2];  // [8,4096,16,16]
  const float* C = (const float*)d_in[3];  // [8,4096,16,16]
  float* Y = (float*)d_out;                // [8,4096,16,64]

  float* wsS = (float*)d_ws;               // chunk states: 8192*1024 floats (32 MB)
  float* wsA = wsS + (size_t)TILES * STATE_ELEMS;  // per-chunk A totals: 8192 floats

  ssd_chunk_kernel<<<TILES, 128, 0, stream>>>(X, A, B, C, Y, wsS, wsA);
  ssd_scan_kernel<<<NB * NH, 256, 0, stream>>>(wsS, wsA);
  ssd_off_kernel<<<TILES, 128, 0, stream>>>(A, C, wsS, Y);
}